// M50_14336600834652
// MI455X (gfx1250) — compile-verified
//
#include <hip/hip_runtime.h>
#include <math.h>

// M50 / EXP-HYDRO neural ODE for MI455X (gfx1250, wave32).
// Kernel 1: sequential RK4 integration, one wave32, lane-per-hidden-neuron MLPs.
//           Weights register-resident; hidden broadcast via v_readlane; forcings in LDS.
// Kernel 2: batched Q-MLP over T rows via V_WMMA_F32_16X16X4_F32 tiles.
// Activations use native gfx1250 TRANS32 ops (v_tanh_f32 / v_exp_f32).

#define T_CONST 1000
#define NSUB 4
#define FMAX_LDS 1024
#define DF_C   2.674f
#define TMAX_C 0.176f
#define TMIN_C (-2.093f)
#define LOG2E_C 1.4426950408889634f

typedef __attribute__((ext_vector_type(2))) float v2f;
typedef __attribute__((ext_vector_type(8))) float v8f;

// ---- native gfx1250 transcendentals (branch-free, single TRANS32 op) ----
__device__ __forceinline__ float fast_tanh(float x) {
#if __has_builtin(__builtin_amdgcn_tanhf)
    return __builtin_amdgcn_tanhf(x);
#elif __has_builtin(__builtin_amdgcn_tanh_f32)
    return __builtin_amdgcn_tanh_f32(x);
#else
    float y;
    asm volatile("v_tanh_f32 %0, %1" : "=v"(y) : "v"(x));
    return y;
#endif
}

__device__ __forceinline__ float fast_exp(float x) {
    // exp(x) = exp2(x * log2(e)); v_exp_f32 saturates correctly at +/-inf
    return __builtin_amdgcn_exp2f(x * LOG2E_C);
}

__device__ __forceinline__ float stepf(float v) {
    return 0.5f * (fast_tanh(5.0f * v) + 1.0f);
}

// Constant-lane broadcast via v_readlane_b32 -> SGPR scalar operand (no LDS).
__device__ __forceinline__ float lane_bcast(float v, int l) {
    return __int_as_float(__builtin_amdgcn_readlane(__float_as_int(v), l));
}

__device__ __forceinline__ float interp1(const float* s, float tt, int Tc) {
    int i0 = (int)floorf(tt);
    if (i0 < 0) i0 = 0;
    if (i0 > Tc - 2) i0 = Tc - 2;
    float fr = tt - (float)i0;
    float a = s[i0], b = s[i0 + 1];
    return a + fr * (b - a);
}

// Per-lane register-resident MLP column (lane j holds column j of every layer).
struct MlpR {
    float w1[3];
    float b1;
    float w2[16];
    float b2;
    float w3;
    float b3;
};

// Cooperative 16-wide MLP: nin -> 16 -> 16 -> 1. All weights in VGPRs; hidden
// activations broadcast across lanes via v_readlane.
__device__ __forceinline__ float mlp_reg(const MlpR& M, const float* xin, int nin) {
    float h = M.b1;
#pragma unroll
    for (int i = 0; i < 3; ++i)
        if (i < nin) h += xin[i] * M.w1[i];
    h = fast_tanh(h);
    float h2 = M.b2;
#pragma unroll
    for (int i = 0; i < 16; ++i) h2 += lane_bcast(h, i) * M.w2[i];
    h2 = fast_tanh(h2);
    float p = h2 * M.w3;
#pragma unroll
    for (int m = 8; m >= 1; m >>= 1) p += __shfl_xor(p, m, 32); // within 16-lane halves
    return p + M.b3;
}

struct Deriv { float d1, d2; };

__device__ __forceinline__ Deriv dyn_eval(float S1, float S2, float tt,
                                          const float* sP, const float* sT,
                                          const float* sL,
                                          const MlpR& E, const MlpR& Q, int Tc) {
    float pr = interp1(sP, tt, Tc);
    float tm = interp1(sT, tt, Tc);
    float ld = interp1(sL, tt, Tc);

    float xin3[3] = {S1, S2, tm};
    float ET = mlp_reg(E, xin3, 3);
    float xin2[3] = {S2, pr, 0.0f};
    float Qo = mlp_reg(Q, xin2, 2);

    float melt = stepf(tm - TMAX_C) * stepf(S1) * fminf(S1, DF_C * (tm - TMAX_C));
    Deriv d;
    d.d1 = stepf(TMIN_C - tm) * pr - melt;
    d.d2 = stepf(tm - TMIN_C) * pr + melt
         - stepf(S2) * ld * fast_exp(ET)
         - stepf(S2) * fast_exp(Qo);
    return d;
}

__global__ __launch_bounds__(32)
void m50_ode_kernel(const float* __restrict__ x,
                    const float* __restrict__ precp, const float* __restrict__ temp,
                    const float* __restrict__ lday,
                    const float* __restrict__ etW1, const float* __restrict__ etb1,
                    const float* __restrict__ etW2, const float* __restrict__ etb2,
                    const float* __restrict__ etW3, const float* __restrict__ etb3,
                    const float* __restrict__ qW1,  const float* __restrict__ qb1,
                    const float* __restrict__ qW2,  const float* __restrict__ qb2,
                    const float* __restrict__ qW3,  const float* __restrict__ qb3,
                    float* __restrict__ s_water, int T) {
    // Stage forcing series in LDS (12 KB of the 320 KB WGP pool) so the serial
    // interpolation chain pays LDS latency, not L2/HBM latency.
    __shared__ float sP[FMAX_LDS], sT[FMAX_LDS], sL[FMAX_LDS];
    const int lane = threadIdx.x;
    const int j    = lane & 15;
    const int Tc   = T < FMAX_LDS ? T : FMAX_LDS;

    for (int i = lane; i < Tc; i += 32) {
        sP[i] = precp[i];
        sT[i] = temp[i];
        sL[i] = lday[i];
    }

    // Load per-lane weight columns into registers (lane j <-> hidden unit j).
    MlpR E, Q;
#pragma unroll
    for (int i = 0; i < 3; ++i) E.w1[i] = etW1[i * 16 + j];
    E.b1 = etb1[j];
#pragma unroll
    for (int i = 0; i < 16; ++i) E.w2[i] = etW2[i * 16 + j];
    E.b2 = etb2[j];
    E.w3 = etW3[j];
    E.b3 = etb3[0];

    Q.w1[0] = qW1[j];
    Q.w1[1] = qW1[16 + j];
    Q.w1[2] = 0.0f;
    Q.b1 = qb1[j];
#pragma unroll
    for (int i = 0; i < 16; ++i) Q.w2[i] = qW2[i * 16 + j];
    Q.b2 = qb2[j];
    Q.w3 = qW3[j];
    Q.b3 = qb3[0];
    __syncthreads();

    float S1 = x[0];   // S_snow(0)  = x[0,0]
    float S2 = x[1];   // S_water(0) = x[0,1]
    if (lane == 0) s_water[0] = S2;

    const float h = 1.0f / (float)NSUB;
    for (int t = 0; t < T - 1; ++t) {
        for (int s = 0; s < NSUB; ++s) {
            float tt = (float)t + (float)s * h;
            Deriv k1 = dyn_eval(S1, S2, tt, sP, sT, sL, E, Q, Tc);
            Deriv k2 = dyn_eval(S1 + 0.5f * h * k1.d1, S2 + 0.5f * h * k1.d2,
                                tt + 0.5f * h, sP, sT, sL, E, Q, Tc);
            Deriv k3 = dyn_eval(S1 + 0.5f * h * k2.d1, S2 + 0.5f * h * k2.d2,
                                tt + 0.5f * h, sP, sT, sL, E, Q, Tc);
            Deriv k4 = dyn_eval(S1 + h * k3.d1, S2 + h * k3.d2,
                                tt + h, sP, sT, sL, E, Q, Tc);
            S1 += h * (k1.d1 + 2.0f * k2.d1 + 2.0f * k3.d1 + k4.d1) * (1.0f / 6.0f);
            S2 += h * (k1.d2 + 2.0f * k2.d2 + 2.0f * k3.d2 + k4.d2) * (1.0f / 6.0f);
        }
        if (lane == 0) s_water[t + 1] = S2;
    }
}

// Batched Q-MLP over T rows using fp32 WMMA 16x16x4 tiles.
// A layout (16x4 f32): lanes 0-15 hold row M=lane {K0,K1}; lanes 16-31 hold {K2,K3}.
// B layout (4x16 f32): lanes 0-15 hold rows K0/K1 at N=lane; lanes 16-31 rows K2/K3.
// C/D layout: VGPR r = (M=r, N=lane) for lanes 0-15; (M=r+8, N=lane-16) for 16-31.
__global__ __launch_bounds__(32)
void m50_qmlp_wmma_kernel(const float* __restrict__ sw, const float* __restrict__ x,
                          const float* __restrict__ qW1, const float* __restrict__ qb1,
                          const float* __restrict__ qW2, const float* __restrict__ qb2,
                          const float* __restrict__ qW3, const float* __restrict__ qb3,
                          float* __restrict__ out, int T) {
    __shared__ float Hld[256];
    const int lane = threadIdx.x;
    const int n    = lane & 15;
    const int half = lane >> 4;
    const int tile = blockIdx.x;

    // ---- Layer 1: [16,2(pad 4)] x [2(pad 4),16] + b1 ----
    const int row  = tile * 16 + n;
    const int rc   = row < T ? row : T - 1;
    const float mk = row < T ? 1.0f : 0.0f;

    v2f a, b;
    a.x = (half == 0) ? sw[rc] * mk         : 0.0f;   // s_water
    a.y = (half == 0) ? x[rc * 3 + 2] * mk  : 0.0f;   // precp = x[:,2]
    b.x = (half == 0) ? qW1[n]      : 0.0f;
    b.y = (half == 0) ? qW1[16 + n] : 0.0f;

    v8f c;
    float bias1 = qb1[n];
#pragma unroll
    for (int r = 0; r < 8; ++r) c[r] = bias1;
    c = __builtin_amdgcn_wmma_f32_16x16x4_f32(false, a, false, b, (short)0, c, false, false);
#pragma unroll
    for (int r = 0; r < 8; ++r) c[r] = fast_tanh(c[r]);

    // Spill H (C-layout) to LDS in plain row-major [m][n] for A-layout reload.
#pragma unroll
    for (int r = 0; r < 8; ++r) Hld[(r + half * 8) * 16 + n] = c[r];
    __syncthreads();

    // ---- Layer 2: [16,16] x [16,16] + b2, as four chained K=4 WMMAs ----
    v8f c2;
    float bias2 = qb2[n];
#pragma unroll
    for (int r = 0; r < 8; ++r) c2[r] = bias2;
#pragma unroll
    for (int k0 = 0; k0 < 16; k0 += 4) {
        const int kb = k0 + half * 2;
        v2f a2, b2v;
        a2.x  = Hld[n * 16 + kb];          // A row M = lane&15
        a2.y  = Hld[n * 16 + kb + 1];
        b2v.x = qW2[kb * 16 + n];
        b2v.y = qW2[(kb + 1) * 16 + n];
        c2 = __builtin_amdgcn_wmma_f32_16x16x4_f32(false, a2, false, b2v, (short)0, c2,
                                                   false, false);
    }
#pragma unroll
    for (int r = 0; r < 8; ++r) c2[r] = fast_tanh(c2[r]);

    // ---- Layer 3: dot with W3 (per-N weight), reduce across 16-lane halves, exp ----
    const float w3    = qW3[n];
    const float bias3 = qb3[0];
#pragma unroll
    for (int r = 0; r < 8; ++r) {
        float p = c2[r] * w3;
#pragma unroll
        for (int m = 8; m >= 1; m >>= 1) p += __shfl_xor(p, m, 32);
        const int orow = tile * 16 + r + half * 8;
        if (n == 0 && orow < T) out[orow] = fast_exp(p + bias3);
    }
}

extern "C" void kernel_launch(void* const* d_in, const int* in_sizes, int n_in,
                              void* d_out, int out_size, void* d_ws, size_t ws_size,
                              hipStream_t stream) {
    const float* x     = (const float*)d_in[0];
    const float* precp = (const float*)d_in[3];
    const float* temp  = (const float*)d_in[4];
    const float* lday  = (const float*)d_in[5];

    const float *etW1, *etb1, *etW2, *etb2, *etW3, *etb3;
    const float *qW1, *qb1, *qW2, *qb2, *qW3, *qb3;
    if (n_in >= 18) {
        // pytree leaves flattened individually
        etW1 = (const float*)d_in[6];  etb1 = (const float*)d_in[7];
        etW2 = (const float*)d_in[8];  etb2 = (const float*)d_in[9];
        etW3 = (const float*)d_in[10]; etb3 = (const float*)d_in[11];
        qW1  = (const float*)d_in[12]; qb1  = (const float*)d_in[13];
        qW2  = (const float*)d_in[14]; qb2  = (const float*)d_in[15];
        qW3  = (const float*)d_in[16]; qb3  = (const float*)d_in[17];
    } else {
        // tuples concatenated into one blob each
        const float* et = (const float*)d_in[6];
        const float* q  = (const float*)d_in[7];
        etW1 = et;       etb1 = et + 48;  etW2 = et + 64;
        etb2 = et + 320; etW3 = et + 336; etb3 = et + 352;
        qW1  = q;        qb1  = q + 32;   qW2  = q + 48;
        qb2  = q + 304;  qW3  = q + 320;  qb3  = q + 336;
    }

    const int T = in_sizes[3] > 0 ? in_sizes[3] : T_CONST;
    float* s_water = (float*)d_ws;     // [T] trajectory scratch
    float* out     = (float*)d_out;    // [T,1]

    // Sequential neural-ODE integration on a single wave32.
    m50_ode_kernel<<<1, 32, 0, stream>>>(x, precp, temp, lday,
                                         etW1, etb1, etW2, etb2, etW3, etb3,
                                         qW1, qb1, qW2, qb2, qW3, qb3,
                                         s_water, T);

    // Batched discharge MLP via fp32 WMMA tiles (in-order on stream).
    const int tiles = (T + 15) / 16;
    m50_qmlp_wmma_kernel<<<tiles, 32, 0, stream>>>(s_water, x,
                                                   qW1, qb1, qW2, qb2, qW3, qb3,
                                                   out, T);
}